// algorithm_37752762532414
// MI455X (gfx1250) — compile-verified
//
#include <hip/hip_runtime.h>

// out = clip(x + laplace_noise * mask, 0, 1)  -- elementwise, fp32.
// Memory-bound streaming kernel: 16 B/elem, ~616 MB total, HBM roofline ~26us.
// Strategy: 128-bit vector loads/stores with non-temporal hints (single-pass
// data, working set >> 192MB L2), v_fmed3 clamp, wave32 256-thread blocks.

typedef float v4f __attribute__((ext_vector_type(4)));

__global__ __launch_bounds__(256) void rrn_clip_vec4(
    const v4f* __restrict__ x,
    const v4f* __restrict__ noise,
    const v4f* __restrict__ mask,
    v4f* __restrict__ out,
    int n4)
{
    int i = blockIdx.x * blockDim.x + threadIdx.x;
    if (i >= n4) return;

    // NT loads: th:TH_LOAD_NT -> don't displace L2 lines; data touched once.
    v4f xv = __builtin_nontemporal_load(x + i);
    v4f nv = __builtin_nontemporal_load(noise + i);
    v4f mv = __builtin_nontemporal_load(mask + i);

    v4f r;
#pragma unroll
    for (int k = 0; k < 4; ++k) {
        float v = __builtin_fmaf(nv[k], mv[k], xv[k]);    // v_fma_f32
        r[k] = __builtin_amdgcn_fmed3f(v, 0.0f, 1.0f);    // v_med3_f32 clamp
    }

    // NT store: write-rinse, don't keep dirty streaming output in cache.
    __builtin_nontemporal_store(r, out + i);
}

// Scalar tail for element counts not divisible by 4 (not hit for 256*3*224*224,
// but keeps the kernel correct for any size).
__global__ __launch_bounds__(256) void rrn_clip_tail(
    const float* __restrict__ x,
    const float* __restrict__ noise,
    const float* __restrict__ mask,
    float* __restrict__ out,
    int base, int n)
{
    int i = base + blockIdx.x * blockDim.x + threadIdx.x;
    if (i >= n) return;
    float v = __builtin_fmaf(noise[i], mask[i], x[i]);
    out[i] = __builtin_amdgcn_fmed3f(v, 0.0f, 1.0f);
}

extern "C" void kernel_launch(void* const* d_in, const int* in_sizes, int n_in,
                              void* d_out, int out_size, void* d_ws, size_t ws_size,
                              hipStream_t stream)
{
    // setup_inputs() order: x, eps (scalar, unused), laplace_noise, mask
    const float* x     = (const float*)d_in[0];
    const float* noise = (const float*)d_in[2];
    const float* mask  = (const float*)d_in[3];
    float* out = (float*)d_out;

    const int n  = in_sizes[0];      // 38,535,168
    const int n4 = n >> 2;           // vec4 elements
    const int rem = n - (n4 << 2);

    if (n4 > 0) {
        const int threads = 256;     // 8 wave32 waves per block
        const int blocks  = (n4 + threads - 1) / threads;
        rrn_clip_vec4<<<blocks, threads, 0, stream>>>(
            (const v4f*)x, (const v4f*)noise, (const v4f*)mask, (v4f*)out, n4);
    }
    if (rem > 0) {
        rrn_clip_tail<<<1, 256, 0, stream>>>(x, noise, mask, out, n4 << 2, n);
    }
}